// TopKMS_36352603193537
// MI455X (gfx1250) — compile-verified
//
#include <hip/hip_runtime.h>

// OHEM top-k MSE loss for MI455X (gfx1250, wave32).
// Stage 1: per-row MSE via V_WMMA_F32_16X16X4_F32 (rowsum = diffsq x ones).
// Stage 2: exact radix-select top-k sum in one workgroup (values >= 0, so
//          uint32 bit order == float order).

typedef __attribute__((ext_vector_type(2))) float v2f;
typedef __attribute__((ext_vector_type(8))) float v8f;

#define D_DIM 512

// One wave reduces a 16-row x 512-col tile: 128 chained WMMAs, C-accumulate.
// A-matrix f32 16x4 layout (ISA 7.12.2): lanes 0-15 hold K=0,1 (VGPR0,1) for
// M=lane; lanes 16-31 hold K=2,3 for M=lane-16. B = all-ones, so every column
// of D holds the 16 row sums regardless of B's lane layout.
__global__ __launch_bounds__(256) void ohem_rowmse_wmma(
    const float* __restrict__ inp, const float* __restrict__ tgt,
    float* __restrict__ per_row)
{
    const int lane  = threadIdx.x & 31;
    const int wave  = threadIdx.x >> 5;
    const int row0  = blockIdx.x * 128 + wave * 16;   // 8 waves * 16 rows
    const int m     = lane & 15;                      // row within tile
    const int khalf = lane >> 4;                      // 0: K=0,1  1: K=2,3

    const size_t base = (size_t)(row0 + m) * D_DIM + (size_t)(khalf * 2);
    const float* pa = inp + base;
    const float* pb = tgt + base;

    v8f acc = {0.f, 0.f, 0.f, 0.f, 0.f, 0.f, 0.f, 0.f};
    const v2f ones = {1.0f, 1.0f};

#pragma unroll 8
    for (int c = 0; c < D_DIM; c += 4) {
        v2f a = *(const v2f*)(pa + c);
        v2f b = *(const v2f*)(pb + c);
        v2f d = a - b;
        v2f sq = d * d;
        // D = A(16x4) * ones(4x16) + C  ->  每 column = row sums
        acc = __builtin_amdgcn_wmma_f32_16x16x4_f32(
            /*neg_a=*/false, sq, /*neg_b=*/false, ones,
            /*c_mod=*/(short)0, acc, /*reuse_a=*/false, /*reuse_b=*/false);
    }

    // C/D layout: VGPR r, lane 0 -> (M=r, N=0); lane 16 -> (M=8+r, N=0).
    if (m == 0) {
        const float scale = 1.0f / (float)D_DIM;
        float* dst = per_row + row0 + khalf * 8;
#pragma unroll
        for (int r = 0; r < 8; ++r) dst[r] = acc[r] * scale;
    }
}

// Single-workgroup exact top-k mean. 4-pass byte radix select over the uint32
// bit patterns (all values >= 0), then sum of values strictly greater than the
// k-th-largest plus the residual tie count at the threshold.
__global__ __launch_bounds__(1024) void ohem_topk_mean(
    const float* __restrict__ per_row, float* __restrict__ out,
    int n, int kk)
{
    __shared__ unsigned int hist[256];
    __shared__ unsigned int s_prefix;
    __shared__ unsigned int s_rank;
    __shared__ float        red[1024];

    const int tid = threadIdx.x;
    if (tid == 0) { s_prefix = 0u; s_rank = (unsigned)kk; }
    __syncthreads();

    for (int pass = 0; pass < 4; ++pass) {
        const int shift = 24 - pass * 8;
        if (tid < 256) hist[tid] = 0u;
        __syncthreads();

        const unsigned prefix = s_prefix;
        const unsigned pmask  = (pass == 0) ? 0u
                                            : (0xFFFFFFFFu << (shift + 8));
        for (int i = tid; i < n; i += 1024) {
            unsigned u = __float_as_uint(per_row[i]);
            if ((u & pmask) == prefix)
                atomicAdd(&hist[(u >> shift) & 255u], 1u);
        }
        __syncthreads();

        if (tid == 0) {
            unsigned rank = s_rank;
            unsigned cum  = 0u;
            int b = 255;
            for (; b > 0; --b) {             // find bucket of k-th largest
                unsigned h = hist[b];
                if (cum + h >= rank) break;
                cum += h;
            }
            s_prefix = prefix | ((unsigned)b << shift);
            s_rank   = rank - cum;           // rank within chosen bucket
        }
        __syncthreads();
    }

    const unsigned T    = s_prefix;          // bit pattern of k-th largest
    const float    tval = __uint_as_float(T);
    const unsigned n_eq = s_rank;            // # copies of T inside top-k

    float local = 0.0f;
    for (int i = tid; i < n; i += 1024) {
        float v = per_row[i];
        if (__float_as_uint(v) > T) local += v;
    }
    red[tid] = local;
    __syncthreads();
#pragma unroll
    for (int s = 512; s > 0; s >>= 1) {
        if (tid < s) red[tid] += red[tid + s];
        __syncthreads();
    }
    if (tid == 0) out[0] = (red[0] + (float)n_eq * tval) / (float)kk;
}

extern "C" void kernel_launch(void* const* d_in, const int* in_sizes, int n_in,
                              void* d_out, int out_size, void* d_ws, size_t ws_size,
                              hipStream_t stream)
{
    (void)n_in; (void)out_size; (void)ws_size;
    const float* inp = (const float*)d_in[0];
    const float* tgt = (const float*)d_in[1];
    float* per_row   = (float*)d_ws;          // n floats (256 KB)
    float* out       = (float*)d_out;

    const int n  = in_sizes[0] / D_DIM;       // 65536
    const int kk = (int)(0.3 * (double)n);    // int(K_FRAC * n) = 19660

    const int blocks = n / 128;               // 16 rows/wave * 8 waves/block
    hipLaunchKernelGGL(ohem_rowmse_wmma, dim3(blocks), dim3(256), 0, stream,
                       inp, tgt, per_row);
    hipLaunchKernelGGL(ohem_topk_mean, dim3(1), dim3(1024), 0, stream,
                       per_row, out, n, kk);
}